// OptimizedTemporalGATLayer_85048942396183
// MI455X (gfx1250) — compile-verified
//
#include <hip/hip_runtime.h>
#include <math.h>

typedef float v2f __attribute__((ext_vector_type(2)));
typedef float v4f __attribute__((ext_vector_type(4)));
typedef float v8f __attribute__((ext_vector_type(8)));

#define HEADS 4
#define DHEAD 64
#define CDIM  256
#define INDIM 192
#define TDIM  64

// ---- order-preserving float<->uint encoding for atomic segment-max ----
__device__ __forceinline__ unsigned encOrd(float f) {
  unsigned u = __float_as_uint(f);
  return (u & 0x80000000u) ? ~u : (u | 0x80000000u);
}
__device__ __forceinline__ float decOrd(unsigned u) {
  unsigned b = (u & 0x80000000u) ? (u & 0x7fffffffu) : ~u;
  return __uint_as_float(b);
}

// encOrd(-inf) == 0x007FFFFF
__global__ void k_init_emax(unsigned* __restrict__ emax, int n) {
  int i = blockIdx.x * blockDim.x + threadIdx.x;
  if (i < n) emax[i] = 0x007FFFFFu;
}

// time embedding cos(t*w+b), scatter mean-agg numerator into combined[:,192:256], degree count
__global__ void k_time_agg(const float* __restrict__ tt, const int* __restrict__ dst,
                           const float* __restrict__ tw, const float* __restrict__ tb,
                           float* __restrict__ combined, float* __restrict__ deg, int E) {
  int e = blockIdx.x * 4 + (threadIdx.x >> 6);
  int t = threadIdx.x & 63;
  if (e >= E) return;
  int d = dst[e];
  float v = cosf(tt[e] * tw[t] + tb[t]);
  unsafeAtomicAdd(&combined[(size_t)d * CDIM + INDIM + t], v);
  if (t == 0) unsafeAtomicAdd(&deg[d], 1.0f);
}

// combined[:,0:192] = h ; combined[:,192:] /= max(deg,1)
__global__ void k_build(const float* __restrict__ h, const float* __restrict__ deg,
                        float* __restrict__ combined, int Nn) {
  size_t i = (size_t)blockIdx.x * blockDim.x + threadIdx.x;
  if (i >= (size_t)Nn * CDIM) return;
  int n = (int)(i >> 8);
  int c = (int)(i & 255);
  if (c < INDIM) combined[i] = h[(size_t)n * INDIM + c];
  else           combined[i] = combined[i] / fmaxf(deg[n], 1.0f);
}

// feat[N,256] = combined[N,256] @ B[256,256]  via V_WMMA_F32_16X16X4_F32
__global__ void __launch_bounds__(256)
k_gemm_wmma(const float* __restrict__ A, const float* __restrict__ B,
            float* __restrict__ C, int Nrows) {
  __shared__ float As[16 * 260];  // pitch 260 -> conflict-free reads
  int rowBase = blockIdx.x * 16;
  int t = threadIdx.x;
#pragma unroll
  for (int it = 0; it < 4; ++it) {
    int flat = it * 1024 + t * 4;
    int r = flat >> 8, c = flat & 255;
    v4f val = {0.f, 0.f, 0.f, 0.f};
    int gr = rowBase + r;
    if (gr < Nrows) val = *(const v4f*)(A + (size_t)gr * CDIM + c);
    *(v4f*)(&As[r * 260 + c]) = val;
  }
  __syncthreads();

  int wave = t >> 5, lane = t & 31;
  int m = lane & 15;
  bool hi = lane >= 16;
  int koff = hi ? 2 : 0;
  int n0 = wave * 16;         // column tiles 0..7
  int n1 = (wave + 8) * 16;   // column tiles 8..15
  v8f acc0 = {};
  v8f acc1 = {};
  for (int k0 = 0; k0 < 256; k0 += 4) {
    int ka = k0 + koff;
    v2f a;
    a.x = As[m * 260 + ka];
    a.y = As[m * 260 + ka + 1];
    const float* B0 = B + (size_t)ka * CDIM;
    __builtin_prefetch(B0 + 4 * CDIM, 0, 1);   // global_prefetch_b8: next K panel
    v2f b0, b1;
    b0.x = B0[n0 + m];        b0.y = B0[CDIM + n0 + m];
    b1.x = B0[n1 + m];        b1.y = B0[CDIM + n1 + m];
    acc0 = __builtin_amdgcn_wmma_f32_16x16x4_f32(false, a, false, b0, (short)0, acc0, false, false);
    acc1 = __builtin_amdgcn_wmma_f32_16x16x4_f32(false, a, false, b1, (short)0, acc1, false, false);
  }
  int cc = lane & 15;
  int rb = rowBase + (hi ? 8 : 0);
#pragma unroll
  for (int v = 0; v < 8; ++v) {
    int gr = rb + v;
    if (gr < Nrows) {
      C[(size_t)gr * CDIM + n0 + cc] = acc0[v];
      C[(size_t)gr * CDIM + n1 + cc] = acc1[v];
    }
  }
}

// el/er: per (node,head) dot of feat[n,h,:] with attn vectors, wave32 reduction
__global__ void k_attn(const float* __restrict__ feat, const float* __restrict__ al,
                       const float* __restrict__ ar, float* __restrict__ el,
                       float* __restrict__ er, int NH) {
  int wid = blockIdx.x * 8 + (threadIdx.x >> 5);
  int lane = threadIdx.x & 31;
  if (wid >= NH) return;
  int n = wid >> 2, hh = wid & 3;
  const float* f   = feat + (size_t)n * CDIM + hh * DHEAD;
  const float* alh = al + hh * DHEAD;
  const float* arh = ar + hh * DHEAD;
  float f0 = f[lane], f1 = f[lane + 32];
  float sl = f0 * alh[lane] + f1 * alh[lane + 32];
  float sr = f0 * arh[lane] + f1 * arh[lane + 32];
  for (int off = 16; off > 0; off >>= 1) {
    sl += __shfl_xor(sl, off);
    sr += __shfl_xor(sr, off);
  }
  if (lane == 0) { el[wid] = sl; er[wid] = sr; }
}

// edge logits + leaky_relu + atomic segment max over dst
__global__ void k_edge_logit(const int* __restrict__ src, const int* __restrict__ dst,
                             const float* __restrict__ el, const float* __restrict__ er,
                             float* __restrict__ exw, unsigned* __restrict__ emax, int E) {
  int i = blockIdx.x * blockDim.x + threadIdx.x;
  if (i >= E * HEADS) return;
  int e = i >> 2, hh = i & 3;
  float v = el[src[e] * HEADS + hh] + er[dst[e] * HEADS + hh];
  v = v > 0.f ? v : 0.01f * v;
  exw[i] = v;
  atomicMax(&emax[dst[e] * HEADS + hh], encOrd(v));
}

// ex = exp(e - max), denominator segment-sum
__global__ void k_edge_exp(const int* __restrict__ dst, const unsigned* __restrict__ emax,
                           float* __restrict__ exw, float* __restrict__ denom, int E) {
  int i = blockIdx.x * blockDim.x + threadIdx.x;
  if (i >= E * HEADS) return;
  int e = i >> 2, hh = i & 3;
  int d = dst[e];
  float m = decOrd(emax[d * HEADS + hh]);
  if (!isfinite(m)) m = 0.f;
  float ex = expf(exw[i] - m);
  exw[i] = ex;
  unsafeAtomicAdd(&denom[d * HEADS + hh], ex);
}

// msg = feat[src]*a, scatter-add to out[dst]; one block per edge, 256 channels
__global__ void k_message(const int* __restrict__ src, const int* __restrict__ dst,
                          const float* __restrict__ feat, const float* __restrict__ exw,
                          const float* __restrict__ denom, float* __restrict__ out, int E) {
  int e = blockIdx.x;
  int c = threadIdx.x;
  int s = src[e], d = dst[e];
  int hh = c >> 6;
  float a = exw[e * HEADS + hh] / fmaxf(denom[d * HEADS + hh], 1e-16f);
  unsafeAtomicAdd(&out[(size_t)d * CDIM + c], feat[(size_t)s * CDIM + c] * a);
}

// residual + ELU
__global__ void k_final(const float* __restrict__ combined, float* __restrict__ out, size_t total) {
  size_t i = (size_t)blockIdx.x * blockDim.x + threadIdx.x;
  if (i >= total) return;
  float x = out[i] + combined[i];
  out[i] = x > 0.f ? x : expm1f(x);
}

extern "C" void kernel_launch(void* const* d_in, const int* in_sizes, int n_in,
                              void* d_out, int out_size, void* d_ws, size_t ws_size,
                              hipStream_t stream) {
  const float* h   = (const float*)d_in[0];
  const float* tt  = (const float*)d_in[1];
  const int*   src = (const int*)d_in[2];
  const int*   dst = (const int*)d_in[3];
  const float* tw  = (const float*)d_in[4];
  const float* tb  = (const float*)d_in[5];
  const float* fcw = (const float*)d_in[6];
  const float* al  = (const float*)d_in[7];
  const float* ar  = (const float*)d_in[8];
  int Nn = in_sizes[0] / INDIM;
  int E  = in_sizes[1];
  float* out = (float*)d_out;

  char* ws = (char*)d_ws;
  size_t off = 0;
  auto alloc = [&](size_t bytes) -> void* {
    void* p = ws + off;
    off += bytes;
    off = (off + 255) & ~(size_t)255;
    return p;
  };
  float*    combined = (float*)alloc((size_t)Nn * CDIM * 4);
  float*    feat     = (float*)alloc((size_t)Nn * CDIM * 4);
  float*    deg      = (float*)alloc((size_t)Nn * 4);
  float*    el       = (float*)alloc((size_t)Nn * HEADS * 4);
  float*    er       = (float*)alloc((size_t)Nn * HEADS * 4);
  unsigned* emax     = (unsigned*)alloc((size_t)Nn * HEADS * 4);
  float*    denom    = (float*)alloc((size_t)Nn * HEADS * 4);
  float*    exw      = (float*)alloc((size_t)E * HEADS * 4);

  hipMemsetAsync(combined, 0, (size_t)Nn * CDIM * 4, stream);
  hipMemsetAsync(deg,      0, (size_t)Nn * 4, stream);
  hipMemsetAsync(denom,    0, (size_t)Nn * HEADS * 4, stream);
  hipMemsetAsync(out,      0, (size_t)out_size * 4, stream);

  k_init_emax <<<(Nn * HEADS + 255) / 256, 256, 0, stream>>>(emax, Nn * HEADS);
  k_time_agg  <<<(E + 3) / 4,             256, 0, stream>>>(tt, dst, tw, tb, combined, deg, E);
  k_build     <<<Nn,                      256, 0, stream>>>(h, deg, combined, Nn);
  k_gemm_wmma <<<(Nn + 15) / 16,          256, 0, stream>>>(combined, fcw, feat, Nn);
  k_attn      <<<(Nn * HEADS + 7) / 8,    256, 0, stream>>>(feat, al, ar, el, er, Nn * HEADS);
  k_edge_logit<<<(E * HEADS + 255) / 256, 256, 0, stream>>>(src, dst, el, er, exw, emax, E);
  k_edge_exp  <<<(E * HEADS + 255) / 256, 256, 0, stream>>>(dst, emax, exw, denom, E);
  k_message   <<<E,                       256, 0, stream>>>(src, dst, feat, exw, denom, out, E);
  k_final     <<<Nn,                      256, 0, stream>>>(combined, out, (size_t)Nn * CDIM);
}